// NNUEHalfKP_49306224558639
// MI455X (gfx1250) — compile-verified
//
#include <hip/hip_runtime.h>
#include <hip/hip_bf16.h>

typedef __attribute__((ext_vector_type(2))) float v2f;
typedef __attribute__((ext_vector_type(8))) float v8f;

#define HIDDEN 256
#define FC2DIM 32

__device__ __forceinline__ void f4acc(float4& a, const float4 b) {
    a.x += b.x; a.y += b.y; a.z += b.z; a.w += b.w;
}

// ---------------------------------------------------------------------------
// Kernel 1: EmbeddingBag sum + bias + ReLU.  One wave32 per bag.
// Each lane owns 8 of the 256 columns as two float4 slices:
//   cols [lane*4, lane*4+4) and [128+lane*4, 128+lane*4+4)  -> b128 loads,
// perfectly coalesced across the wave.  2-way unrolled over bag indices so
// four global_load_b128 are in flight before the first s_wait_loadcnt.
// ---------------------------------------------------------------------------
__global__ __launch_bounds__(256) void embed_bag_relu_kernel(
    const int* __restrict__ indices, const int* __restrict__ offsets,
    const float* __restrict__ ew, const float* __restrict__ bias1,
    float* __restrict__ h, int batch, int n_idx)
{
    const int bag  = (blockIdx.x * blockDim.x + threadIdx.x) >> 5;
    const int lane = threadIdx.x & 31;
    if (bag >= batch) return;

    const int start = offsets[bag];
    const int end   = (bag + 1 < batch) ? offsets[bag + 1] : n_idx;

    float4 acc0 = make_float4(0.f, 0.f, 0.f, 0.f);
    float4 acc1 = make_float4(0.f, 0.f, 0.f, 0.f);

    int i = start;
    for (; i + 1 < end; i += 2) {
        const float4* r0 = (const float4*)(ew + (size_t)indices[i]     * HIDDEN);
        const float4* r1 = (const float4*)(ew + (size_t)indices[i + 1] * HIDDEN);
        float4 x0 = r0[lane];
        float4 x1 = r0[lane + 32];
        float4 y0 = r1[lane];
        float4 y1 = r1[lane + 32];
        f4acc(acc0, x0); f4acc(acc1, x1);
        f4acc(acc0, y0); f4acc(acc1, y1);
    }
    if (i < end) {
        const float4* r0 = (const float4*)(ew + (size_t)indices[i] * HIDDEN);
        f4acc(acc0, r0[lane]);
        f4acc(acc1, r0[lane + 32]);
    }

    const float4 b0 = ((const float4*)bias1)[lane];
    const float4 b1 = ((const float4*)bias1)[lane + 32];
    float4 o0, o1;
    o0.x = fmaxf(acc0.x + b0.x, 0.f); o0.y = fmaxf(acc0.y + b0.y, 0.f);
    o0.z = fmaxf(acc0.z + b0.z, 0.f); o0.w = fmaxf(acc0.w + b0.w, 0.f);
    o1.x = fmaxf(acc1.x + b1.x, 0.f); o1.y = fmaxf(acc1.y + b1.y, 0.f);
    o1.z = fmaxf(acc1.z + b1.z, 0.f); o1.w = fmaxf(acc1.w + b1.w, 0.f);

    float4* hp = (float4*)(h + (size_t)bag * HIDDEN);
    hp[lane]      = o0;
    hp[lane + 32] = o1;
}

// ---------------------------------------------------------------------------
// Kernel 2: fused MLP head via f32 WMMA.  One wave32 per 16-row batch tile.
//   D = h[16,256] x fc2_w^T[256,32]  using V_WMMA_F32_16X16X4_F32,
//   64 K-steps, two independent accumulator chains (N-tile 0 and 1).
// VGPR layouts per CDNA5 ISA 7.12.2:
//   A 16x4 f32 : lane<16 -> M=lane, {K,K+1}; lane>=16 -> M=lane-16, {K+2,K+3}
//   B 4x16 f32 : lane<16 -> N=lane, rows {K,K+1}; lane>=16 -> rows {K+2,K+3}
//   C/D 16x16  : vgpr j, lanes 0-15 -> M=j, N=lane; lanes 16-31 -> M=j+8.
// Then fc2 bias+ReLU and the 32->1 output layer fused as per-lane FMA with a
// 16-lane shfl_xor reduction (halves of the wave reduce rows j and j+8).
// ---------------------------------------------------------------------------
__global__ __launch_bounds__(256) void mlp_wmma_kernel(
    const float* __restrict__ h, const float* __restrict__ fc2_w,
    const float* __restrict__ fc2_b, const float* __restrict__ out_w,
    const float* __restrict__ out_b, float* __restrict__ out, int batch)
{
    const int wave = (blockIdx.x * blockDim.x + threadIdx.x) >> 5;
    const int lane = threadIdx.x & 31;
    const int ntiles = batch >> 4;
    if (wave >= ntiles) return;

    const int m0   = wave << 4;
    const int half = lane >> 4;        // 0: K pair {0,1}, 1: K pair {2,3}
    const int r    = lane & 15;        // M for A, N for B/C

    const float* Ab  = h     + (size_t)(m0 + r)  * HIDDEN + 2 * half;
    const float* B0b = fc2_w + (size_t)r         * HIDDEN + 2 * half;
    const float* B1b = fc2_w + (size_t)(16 + r)  * HIDDEN + 2 * half;

    v8f c0 = {};
    v8f c1 = {};

#pragma unroll 4
    for (int k = 0; k < HIDDEN; k += 4) {
        v2f a  = *(const v2f*)(Ab  + k);
        v2f b0 = *(const v2f*)(B0b + k);
        v2f b1 = *(const v2f*)(B1b + k);
        // 8 args: (neg_a, A, neg_b, B, c_mod, C, reuse_a, reuse_b)
        c0 = __builtin_amdgcn_wmma_f32_16x16x4_f32(
                false, a, false, b0, (short)0, c0, false, false);
        c1 = __builtin_amdgcn_wmma_f32_16x16x4_f32(
                false, a, false, b1, (short)0, c1, false, false);
    }

    const float fb0 = fc2_b[r];
    const float fb1 = fc2_b[16 + r];
    const float ow0 = out_w[r];
    const float ow1 = out_w[16 + r];
    const float ob  = out_b[0];

#pragma unroll
    for (int j = 0; j < 8; ++j) {
        float h0 = fmaxf(c0[j] + fb0, 0.f);
        float h1 = fmaxf(c1[j] + fb1, 0.f);
        float p  = h0 * ow0 + h1 * ow1;
        // reduce across the 16-lane half (xor masks < 16 never cross halves)
        p += __shfl_xor(p, 8, 32);
        p += __shfl_xor(p, 4, 32);
        p += __shfl_xor(p, 2, 32);
        p += __shfl_xor(p, 1, 32);
        if (r == 0) out[m0 + j + 8 * half] = p + ob;
    }
}

// ---------------------------------------------------------------------------
// Host launcher.
// Inputs (setup_inputs order): indices i32[N], offsets i32[B],
//   embed_weight f32[41024*256], bias1 f32[256], fc2_w f32[32*256],
//   fc2_b f32[32], out_w f32[32], out_b f32[1].
// Output: f32[B].  Workspace: h buffer f32[B*256] = 16.8 MB.
// ---------------------------------------------------------------------------
extern "C" void kernel_launch(void* const* d_in, const int* in_sizes, int n_in,
                              void* d_out, int out_size, void* d_ws, size_t ws_size,
                              hipStream_t stream) {
    const int*   indices = (const int*)  d_in[0];
    const int*   offsets = (const int*)  d_in[1];
    const float* ew      = (const float*)d_in[2];
    const float* bias1   = (const float*)d_in[3];
    const float* fc2_w   = (const float*)d_in[4];
    const float* fc2_b   = (const float*)d_in[5];
    const float* out_w   = (const float*)d_in[6];
    const float* out_b   = (const float*)d_in[7];
    float*       out     = (float*)d_out;
    float*       hbuf    = (float*)d_ws;

    const int n_idx = in_sizes[0];
    const int batch = in_sizes[1];

    // Kernel 1: one wave per bag, 8 waves (256 threads) per block.
    const int blocks1 = (batch * 32 + 255) / 256;
    embed_bag_relu_kernel<<<blocks1, 256, 0, stream>>>(
        indices, offsets, ew, bias1, hbuf, batch, n_idx);

    // Kernel 2: one wave per 16-row tile.
    const int ntiles  = batch / 16;
    const int blocks2 = (ntiles * 32 + 255) / 256;
    mlp_wmma_kernel<<<blocks2, 256, 0, stream>>>(
        hbuf, fc2_w, fc2_b, out_w, out_b, out, batch);
}